// HybridModel_61976378082038
// MI455X (gfx1250) — compile-verified
//
#include <hip/hip_runtime.h>
#include <hip/hip_bf16.h>
#include <math.h>

typedef __attribute__((ext_vector_type(2))) float v2f;
typedef __attribute__((ext_vector_type(8))) float v8f;

#define N_IMG   8192
#define PPI     196          // patches per image (14*14)
#define WAVES   13           // ceil(196/16)
#define BLOCK   (WAVES * 32) // 416 threads

// ---------------------------------------------------------------------------
// Kernel 1: build fixed 16x16 complex unitary U from qparams.
// Lane j propagates basis state |j> through: RX(q0,w0) RY(q1,w1) CNOT(0,1)
// RZ(q2,w2) CNOT(2,3) RX(q3,w3) RY(q4,w2) CNOT(1,3).
// wire w <-> bit (3-w) of the flat amplitude index.
// Output V: rows 0..15 = Re(U), rows 16..31 = Im(U), row-major [32][16].
// ---------------------------------------------------------------------------
__global__ void build_unitary_kernel(const float* __restrict__ qp,
                                     float* __restrict__ V) {
    const int j = threadIdx.x;
    if (j >= 16) return;
    float sr[16], si[16];
    #pragma unroll
    for (int i = 0; i < 16; ++i) { sr[i] = (i == j) ? 1.f : 0.f; si[i] = 0.f; }

    auto ap1q = [&](int w, float m00r, float m00i, float m01r, float m01i,
                            float m10r, float m10i, float m11r, float m11i) {
        const int msk = 8 >> w;
        #pragma unroll
        for (int i = 0; i < 16; ++i) {
            if (!(i & msk)) {
                const int k = i | msk;
                float ar = sr[i], ai = si[i], br = sr[k], bi = si[k];
                sr[i] = m00r*ar - m00i*ai + m01r*br - m01i*bi;
                si[i] = m00r*ai + m00i*ar + m01r*bi + m01i*br;
                sr[k] = m10r*ar - m10i*ai + m11r*br - m11i*bi;
                si[k] = m10r*ai + m10i*ar + m11r*bi + m11i*br;
            }
        }
    };
    auto cnot = [&](int c, int t) {
        const int cm = 8 >> c, tm = 8 >> t;
        #pragma unroll
        for (int i = 0; i < 16; ++i) {
            if ((i & cm) && !(i & tm)) {
                const int k = i | tm;
                float tr = sr[i], ti = si[i];
                sr[i] = sr[k]; si[i] = si[k]; sr[k] = tr; si[k] = ti;
            }
        }
    };

    float t, c, s;
    t = qp[0]; c = cosf(0.5f*t); s = sinf(0.5f*t);           // RX wire0
    ap1q(0, c,0.f, 0.f,-s, 0.f,-s, c,0.f);
    t = qp[1]; c = cosf(0.5f*t); s = sinf(0.5f*t);           // RY wire1
    ap1q(1, c,0.f, -s,0.f, s,0.f, c,0.f);
    cnot(0, 1);
    t = qp[2]; c = cosf(0.5f*t); s = sinf(0.5f*t);           // RZ wire2
    ap1q(2, c,-s, 0.f,0.f, 0.f,0.f, c,s);
    cnot(2, 3);
    t = qp[3]; c = cosf(0.5f*t); s = sinf(0.5f*t);           // RX wire3
    ap1q(3, c,0.f, 0.f,-s, 0.f,-s, c,0.f);
    t = qp[4]; c = cosf(0.5f*t); s = sinf(0.5f*t);           // RY wire2
    ap1q(2, c,0.f, -s,0.f, s,0.f, c,0.f);
    cnot(1, 3);

    #pragma unroll
    for (int i = 0; i < 16; ++i) {
        V[i * 16 + j]        = sr[i];
        V[(16 + i) * 16 + j] = si[i];
    }
}

// ---------------------------------------------------------------------------
// Kernel 2: fused quanvolution + linear + log_softmax. One block per image.
// Each wave handles a 16-patch tile; F_re = U_re*S and F_im = U_im*S via
// 4 chained V_WMMA_F32_16X16X4_F32 each (K=16 in K=4 chunks).
// ---------------------------------------------------------------------------
__global__ __launch_bounds__(BLOCK) void quanv_fused_kernel(
    const float* __restrict__ x,     // [8192, 784]
    const float* __restrict__ V,     // [32, 16] (Re;Im of U)
    const float* __restrict__ W,     // [2, 784]
    const float* __restrict__ bias,  // [2]
    float* __restrict__ out) {       // [8192, 2]
    __shared__ float lacc[2];
    const int img  = blockIdx.x;
    const int tid  = threadIdx.x;
    if (tid == 0) { lacc[0] = 0.f; lacc[1] = 0.f; }
    __syncthreads();

    const int wave = tid >> 5;
    const int lane = tid & 31;
    const int n    = lane & 15;   // patch within tile / matrix column
    const int hi   = lane >> 4;   // lane half selects K pair / row half
    const int p    = wave * 16 + n;            // patch-local id 0..207
    const int pc   = (p < PPI) ? p : (PPI - 1);
    const float validf = (p < PPI) ? 1.f : 0.f;

    // --- load 2x2 patch (angles); masked lanes read clamped addr, angle=0 ---
    const int rh = pc / 14, ch = pc % 14;
    const float* xb = x + img * 784 + rh * 56 + ch * 2;
    const float t0 = xb[0]  * validf;
    const float t1 = xb[1]  * validf;
    const float t2 = xb[28] * validf;
    const float t3 = xb[29] * validf;

    // --- encoder product state s[i] = f0(i>>3&1) f1(i>>2&1) f2(i>>1&1) f3(i&1)
    const float c0 = __cosf(0.5f*t0), s0 = __sinf(0.5f*t0);
    const float c1 = __cosf(0.5f*t1), s1 = __sinf(0.5f*t1);
    const float c2 = __cosf(0.5f*t2), s2 = __sinf(0.5f*t2);
    const float c3 = __cosf(0.5f*t3), s3 = __sinf(0.5f*t3);
    float e01[4];
    e01[0] = c0*c1; e01[1] = c0*s1; e01[2] = s0*c1; e01[3] = s0*s1;
    const float f2   = hi ? s2 : c2;     // this lane-half's i2 bit
    const float e23a = f2 * c3;          // i3 = 0
    const float e23b = f2 * s3;          // i3 = 1

    // --- build WMMA operands: A = U chunks, B = state chunks ---------------
    const int row = n;        // A-matrix row M = lane%16
    const int kb  = hi * 2;   // K base within each K=4 chunk
    v2f Are[4], Aim[4], Bm[4];
    #pragma unroll
    for (int j = 0; j < 4; ++j) {
        const int k = 4 * j + kb;
        Are[j] = (v2f){ V[row * 16 + k],        V[row * 16 + k + 1] };
        Aim[j] = (v2f){ V[(16 + row) * 16 + k], V[(16 + row) * 16 + k + 1] };
        Bm[j]  = (v2f){ e01[j] * e23a,          e01[j] * e23b };
    }

    v8f Fre = {0.f,0.f,0.f,0.f,0.f,0.f,0.f,0.f};
    v8f Fim = {0.f,0.f,0.f,0.f,0.f,0.f,0.f,0.f};
    #pragma unroll
    for (int j = 0; j < 4; ++j) {
        Fre = __builtin_amdgcn_wmma_f32_16x16x4_f32(
                  false, Are[j], false, Bm[j], (short)0, Fre, false, false);
        Fim = __builtin_amdgcn_wmma_f32_16x16x4_f32(
                  false, Aim[j], false, Bm[j], (short)0, Fim, false, false);
    }

    // --- Z expectations: exps[w] = sum_i z_w(i) * |final_i|^2 --------------
    // D layout: VGPR r holds row i = r + 8*hi of column n.
    float ex[4] = {0.f, 0.f, 0.f, 0.f};
    #pragma unroll
    for (int r = 0; r < 8; ++r) {
        const float pr = Fre[r]*Fre[r] + Fim[r]*Fim[r];
        const int i = r + 8 * hi;
        #pragma unroll
        for (int w = 0; w < 4; ++w) {
            const float sgn = ((i >> (3 - w)) & 1) ? -1.f : 1.f;
            ex[w] += sgn * pr;
        }
    }
    #pragma unroll
    for (int w = 0; w < 4; ++w) ex[w] += __shfl_xor(ex[w], 16, 32);

    // --- linear layer contribution (only low half-lanes, valid patches) ----
    const float mask = (hi == 0) ? validf : 0.f;
    const float* W0 = W + 4 * pc;
    const float* W1 = W + 784 + 4 * pc;
    float l0 = 0.f, l1 = 0.f;
    #pragma unroll
    for (int w = 0; w < 4; ++w) { l0 += W0[w] * ex[w]; l1 += W1[w] * ex[w]; }
    l0 *= mask; l1 *= mask;

    // wave reduction, then LDS accumulate across the 13 waves
    #pragma unroll
    for (int off = 16; off >= 1; off >>= 1) {
        l0 += __shfl_xor(l0, off, 32);
        l1 += __shfl_xor(l1, off, 32);
    }
    if (lane == 0) { atomicAdd(&lacc[0], l0); atomicAdd(&lacc[1], l1); }
    __syncthreads();

    if (tid == 0) {
        const float g0 = lacc[0] + bias[0];
        const float g1 = lacc[1] + bias[1];
        const float m  = fmaxf(g0, g1);
        const float lse = m + __logf(__expf(g0 - m) + __expf(g1 - m));
        out[img * 2 + 0] = g0 - lse;
        out[img * 2 + 1] = g1 - lse;
    }
}

// ---------------------------------------------------------------------------
extern "C" void kernel_launch(void* const* d_in, const int* in_sizes, int n_in,
                              void* d_out, int out_size, void* d_ws, size_t ws_size,
                              hipStream_t stream) {
    const float* x   = (const float*)d_in[0];  // [8192*784]
    const float* qp  = (const float*)d_in[1];  // [5]
    const float* W   = (const float*)d_in[2];  // [2*784]
    const float* b   = (const float*)d_in[3];  // [2]
    float* out = (float*)d_out;                // [8192*2]
    float* V   = (float*)d_ws;                 // [32*16] fixed unitary

    build_unitary_kernel<<<1, 32, 0, stream>>>(qp, V);
    quanv_fused_kernel<<<N_IMG, BLOCK, 0, stream>>>(x, V, W, b, out);
}